// three_gcn_36928128811441
// MI455X (gfx1250) — compile-verified
//
#include <hip/hip_runtime.h>
#include <math.h>

// Problem constants (fixed by the reference)
#define KGAUSS 10
#define OUTF   32
#define COLS   320        // K * OUT
#define GEPS   1e-15f

typedef float v2f __attribute__((ext_vector_type(2)));
typedef float v8f __attribute__((ext_vector_type(8)));

// ---------------------------------------------------------------- zero fill
__global__ void zero_kernel(float* __restrict__ p, long n) {
    long i = (long)blockIdx.x * blockDim.x + threadIdx.x;
    if (i < n) p[i] = 0.0f;
}

// ---------------------------------------------------------------- degree
__global__ void deg_kernel(const int* __restrict__ ei, float* __restrict__ deg, int E) {
    int e = blockIdx.x * blockDim.x + threadIdx.x;
    if (e < E) atomicAdd(&deg[ei[E + e]], 1.0f);   // ei[1][e] = dst
}

// ---------------------------------------------------------------- h = x @ g  (fp32 WMMA 16x16x4)
// x: [n, CIN] row-major, g: [CIN, 320] row-major, h: [n, 320] row-major.
// One wave computes one 16x16 output tile. n % 16 == 0 (50000 = 16*3125),
// 320 = 16*20, CIN % 4 == 0, so no partial tiles and EXEC stays all-1s.
template<int CIN>
__global__ __launch_bounds__(256)
void gemm_h_wmma(const float* __restrict__ x, const float* __restrict__ g,
                 float* __restrict__ h, int totalTiles, int colTiles) {
    int waveId = (int)(((long)blockIdx.x * blockDim.x + threadIdx.x) >> 5);
    if (waveId >= totalTiles) return;
    int lane    = threadIdx.x & 31;
    int rowTile = waveId / colTiles;
    int colTile = waveId - rowTile * colTiles;

    int  m    = lane & 15;       // row (A) / col (B,C,D) within tile
    int  kh   = lane >> 4;       // K-half select for A/B, M-half select for C/D
    long row0 = (long)rowTile * 16;
    int  col0 = colTile * 16;

    v8f c = {};
    #pragma unroll
    for (int kk = 0; kk < CIN; kk += 4) {
        // A 16x4: lanes 0-15 hold K={0,1} of row m, lanes 16-31 hold K={2,3}
        const float* ap = x + (row0 + m) * CIN + (kk + 2 * kh);
        v2f a; a.x = ap[0]; a.y = ap[1];
        // B 4x16: lanes 0-15 hold rows K={0,1} at col m, lanes 16-31 K={2,3}
        const float* bp = g + (long)(kk + 2 * kh) * COLS + (col0 + m);
        v2f b; b.x = bp[0]; b.y = bp[COLS];
        c = __builtin_amdgcn_wmma_f32_16x16x4_f32(false, a, false, b,
                                                  (short)0, c, false, false);
    }
    // C/D: VGPR r holds M = r (lanes 0-15) / M = 8+r (lanes 16-31), N = lane&15
    #pragma unroll
    for (int r = 0; r < 8; ++r)
        h[(row0 + 8 * kh + r) * COLS + (col0 + m)] = c[r];
}

// ---------------------------------------------------------------- edge scatter
// One wave per edge; lane j owns output feature j. Lanes 0..9 compute the 10
// Gaussian weights, broadcast via shfl; gather of h[src] is 128B-coalesced per
// k-step (L2 resident), scatter is a coalesced 32-float atomic burst.
__global__ __launch_bounds__(256)
void edge_kernel(const int* __restrict__ ei, const float* __restrict__ eattr,
                 const float* __restrict__ mu, const float* __restrict__ sigma,
                 const float* __restrict__ h, float* __restrict__ agg, int E) {
    int wave = (int)(((long)blockIdx.x * blockDim.x + threadIdx.x) >> 5);
    int lane = threadIdx.x & 31;
    if (wave >= E) return;

    int src = ei[wave];        // edge_index[0][e]
    int dst = ei[E + wave];    // edge_index[1][e]

    float gval = 0.0f;
    if (lane < KGAUSS) {
        float d0 = eattr[2 * wave + 0] - mu[2 * lane + 0];
        float d1 = eattr[2 * wave + 1] - mu[2 * lane + 1];
        float s0 = sigma[2 * lane + 0];
        float s1 = sigma[2 * lane + 1];
        float q  = d0 * d0 / (GEPS + s0 * s0) + d1 * d1 / (GEPS + s1 * s1);
        gval = __expf(-0.5f * q);
    }

    const float* hrow = h + (long)src * COLS;
    float acc = 0.0f;
    #pragma unroll
    for (int k = 0; k < KGAUSS; ++k) {
        float gk = __shfl(gval, k, 32);
        acc = fmaf(gk, hrow[k * OUTF + lane], acc);
    }
    atomicAdd(&agg[(long)dst * OUTF + lane], acc);
}

// ---------------------------------------------------------------- finalize
// y = elu(agg/max(deg,1) + x @ root + b). One thread per (node, out-feature):
// x reads are wave-uniform (scalar path), root reads coalesced across lanes.
template<int CIN>
__global__ __launch_bounds__(256)
void finalize_kernel(const float* __restrict__ agg, const float* __restrict__ deg,
                     const float* __restrict__ x, const float* __restrict__ root,
                     const float* __restrict__ bias, float* __restrict__ out, int n) {
    long idx = (long)blockIdx.x * blockDim.x + threadIdx.x;
    if (idx >= (long)n * OUTF) return;
    int node = (int)(idx >> 5);
    int j    = (int)(idx & 31);

    float acc = agg[idx] / fmaxf(deg[node], 1.0f);
    const float* xr = x + (long)node * CIN;
    float rsum = 0.0f;
    #pragma unroll 8
    for (int c = 0; c < CIN; ++c)
        rsum = fmaf(xr[c], root[c * OUTF + j], rsum);
    float v = acc + rsum + bias[j];
    out[idx] = (v > 0.0f) ? v : expm1f(v);   // ELU, alpha=1
}

// ---------------------------------------------------------------- launch
extern "C" void kernel_launch(void* const* d_in, const int* in_sizes, int n_in,
                              void* d_out, int out_size, void* d_ws, size_t ws_size,
                              hipStream_t stream) {
    const int IN = 64;
    const int N  = in_sizes[0] / IN;       // 50000
    const int E  = in_sizes[2] / 2;        // 800000

    const float* graph = (const float*)d_in[0];
    const int*   ei    = (const int*)  d_in[1];
    const float* eattr = (const float*)d_in[2];
    const float* G [3] = {(const float*)d_in[3],  (const float*)d_in[8],  (const float*)d_in[13]};
    const float* MU[3] = {(const float*)d_in[4],  (const float*)d_in[9],  (const float*)d_in[14]};
    const float* SG[3] = {(const float*)d_in[5],  (const float*)d_in[10], (const float*)d_in[15]};
    const float* RT[3] = {(const float*)d_in[6],  (const float*)d_in[11], (const float*)d_in[16]};
    const float* BB[3] = {(const float*)d_in[7],  (const float*)d_in[12], (const float*)d_in[17]};
    float* out = (float*)d_out;

    // workspace: h [N,320] | agg [N,32] | deg [N]
    float* h_ws  = (float*)d_ws;
    float* agg   = h_ws + (long)N * COLS;
    float* deg   = agg  + (long)N * OUTF;

    const int TB = 256;

    // degree (shared by all layers)
    {
        long n = N;
        zero_kernel<<<(unsigned)((n + TB - 1) / TB), TB, 0, stream>>>(deg, n);
        deg_kernel<<<(unsigned)((E + TB - 1) / TB), TB, 0, stream>>>(ei, deg, E);
    }

    const int colTiles   = COLS / 16;                 // 20
    const int rowTiles   = (N + 15) / 16;             // 3125
    const int totalTiles = rowTiles * colTiles;
    const unsigned gemmBlocks = (unsigned)(((long)totalTiles * 32 + TB - 1) / TB);
    const unsigned edgeBlocks = (unsigned)(((long)E * 32 + TB - 1) / TB);
    const unsigned finBlocks  = (unsigned)(((long)N * OUTF + TB - 1) / TB);

    const float* xin = graph;
    for (int l = 0; l < 3; ++l) {
        float* yl = out + (long)l * N * OUTF;
        long naeg = (long)N * OUTF;
        zero_kernel<<<(unsigned)((naeg + TB - 1) / TB), TB, 0, stream>>>(agg, naeg);

        if (l == 0)
            gemm_h_wmma<64><<<gemmBlocks, TB, 0, stream>>>(xin, G[l], h_ws, totalTiles, colTiles);
        else
            gemm_h_wmma<32><<<gemmBlocks, TB, 0, stream>>>(xin, G[l], h_ws, totalTiles, colTiles);

        edge_kernel<<<edgeBlocks, TB, 0, stream>>>(ei, eattr, MU[l], SG[l], h_ws, agg, E);

        if (l == 0)
            finalize_kernel<64><<<finBlocks, TB, 0, stream>>>(agg, deg, xin, RT[l], BB[l], yl, N);
        else
            finalize_kernel<32><<<finBlocks, TB, 0, stream>>>(agg, deg, xin, RT[l], BB[l], yl, N);

        xin = yl;   // next layer consumes this layer's ELU output
    }
}